// MPNNs_73787538145745
// MI455X (gfx1250) — compile-verified
//
#include <hip/hip_runtime.h>

typedef __attribute__((ext_vector_type(2))) float v2f;
typedef __attribute__((ext_vector_type(8))) float v8f;
typedef int v4i __attribute__((vector_size(4 * sizeof(int))));

#define D 128
#define LAYERS 3

// ---- gfx1250 async global->LDS copy (ASYNCcnt-tracked), with sync fallback --
#if __has_builtin(__builtin_amdgcn_global_load_async_to_lds_b128) && \
    __has_builtin(__builtin_amdgcn_s_wait_asynccnt)
#define HAVE_ASYNC 1
#define ASYNC_COPY16(gp, lp)                                          \
    __builtin_amdgcn_global_load_async_to_lds_b128(                   \
        (__attribute__((address_space(1))) v4i*)(gp),                 \
        (__attribute__((address_space(3))) v4i*)(lp), 0, 0)
#define WAIT_ASYNC(n) __builtin_amdgcn_s_wait_asynccnt(n)
#else
#define HAVE_ASYNC 0
#define WAIT_ASYNC(n) ((void)0)
#endif

// ---------------------------------------------------------------- utilities
__global__ void zero_kernel(float* __restrict__ p, long n) {
    long i = (long)blockIdx.x * blockDim.x + threadIdx.x;
    long stride = (long)gridDim.x * blockDim.x;
    for (; i < n; i += stride) p[i] = 0.0f;
}

__global__ void deg_kernel(const int* __restrict__ dst, float* __restrict__ deg, int E) {
    int e = blockIdx.x * 256 + threadIdx.x;
    if (e < E) atomicAdd(&deg[dst[e]], 1.0f);
}

__global__ void dinv_kernel(float* __restrict__ deg, int N) {
    int i = blockIdx.x * 256 + threadIdx.x;
    if (i < N) deg[i] = rsqrtf(deg[i] + 1.0f);   // deg includes self-loop (+1)
}

// ---------------------------------------------------------------- WMMA GEMM
// Out[N x (16*NCT)] = A[N x 128] @ W[128 x (16*NCT)]  (+= if ACCUM, +bias if BIAS)
// 128-thread block = 4 waves = 128 output rows (two 64-row tiles, A double-
// buffered via async global->LDS loads so tile-1 loads overlap tile-0 WMMA).
template <int NCT, bool ACCUM, bool BIAS>
__global__ __launch_bounds__(128)
void gemm_kernel(const float* __restrict__ A, const float* __restrict__ W,
                 float* __restrict__ Out, const float* __restrict__ bias, int N) {
    constexpr int NC  = NCT * 16;       // output columns
    constexpr int WPS = 2 * NC + 32;    // pair-row stride; WPS % 64 == 32 -> bank split
    constexpr int AS  = 132;            // A row stride: conflict-free frags, 16B-aligned rows
    __shared__ float Alds[2][64 * AS];
    __shared__ float Wb[64 * WPS];      // (k>>1) pair rows, [2n + (k&1)] interleave

    const int t = threadIdx.x;
    const int base0 = blockIdx.x * 128;

    // stage weight panel (sync): W[k][n] -> Wb[(k>>1)*WPS + 2n + (k&1)]
    for (int idx = t; idx < D * NC; idx += 128) {
        int k = idx / NC, n = idx % NC;
        Wb[(k >> 1) * WPS + 2 * n + (k & 1)] = W[idx];
    }

    // stage both A tiles. OOB rows clamp to N-1 (outputs are row-guarded, and
    // accumulator rows never mix, so the duplicated data is harmless). The
    // clamp keeps the per-wave async-instruction count exactly 16 per tile,
    // which the partial s_wait_asynccnt below relies on (in-order completion).
#if HAVE_ASYNC
    for (int tile = 0; tile < 2; ++tile) {
        float* abuf = Alds[tile];
        const int rbase = base0 + tile * 64;
        for (int q = t; q < 64 * 32; q += 128) {      // 16 async b128 per thread
            int r = q >> 5, j = (q & 31) * 4;
            long srow = rbase + r;
            if (srow >= N) srow = N - 1;
            ASYNC_COPY16(A + srow * D + j, &abuf[r * AS + j]);
        }
    }
#else
    for (int tile = 0; tile < 2; ++tile) {
        float* abuf = Alds[tile];
        const int rbase = base0 + tile * 64;
        for (int idx = t; idx < 64 * D; idx += 128) {
            int r = idx >> 7, c = idx & 127;
            long srow = rbase + r;
            if (srow >= N) srow = N - 1;
            abuf[r * AS + c] = A[srow * D + c];
        }
    }
#endif

    const int wave = t >> 5, lane = t & 31;
    const int half = lane >> 4, l16 = lane & 15;
    const int m = wave * 16 + l16;              // A row of this lane within tile
    const v8f vzero = {0.f, 0.f, 0.f, 0.f, 0.f, 0.f, 0.f, 0.f};

    for (int tile = 0; tile < 2; ++tile) {
        if (tile == 0) { WAIT_ASYNC(16); }      // tile-0 loads done; tile-1 in flight
        else           { WAIT_ASYNC(0);  }
        __syncthreads();

        const int rowbase = base0 + tile * 64;
        if (rowbase < N) {                      // block-uniform: skip empty tail tile
            const float* abuf = Alds[tile];
            v8f acc[NCT];
#pragma unroll
            for (int ct = 0; ct < NCT; ++ct) acc[ct] = vzero;

#pragma unroll 4
            for (int k4 = 0; k4 < 32; ++k4) {   // K = 128 in steps of 4
                const int kk = k4 * 4 + half * 2;
                v2f av;
                av.x = abuf[m * AS + kk];
                av.y = abuf[m * AS + kk + 1];
                const int pr = kk >> 1;
#pragma unroll
                for (int ct = 0; ct < NCT; ++ct) {
                    const int n = ct * 16 + l16;
                    v2f bv;
                    bv.x = Wb[pr * WPS + 2 * n];        // W[kk][n]
                    bv.y = Wb[pr * WPS + 2 * n + 1];    // W[kk+1][n]
                    acc[ct] = __builtin_amdgcn_wmma_f32_16x16x4_f32(
                        false, av, false, bv, (short)0, acc[ct], false, false);
                }
            }

            // store: VGPR r holds row (r + half*8), col = ct*16 + l16
#pragma unroll
            for (int ct = 0; ct < NCT; ++ct) {
                const int col = ct * 16 + l16;
#pragma unroll
                for (int r = 0; r < 8; ++r) {
                    const int row = rowbase + wave * 16 + half * 8 + r;
                    if (row < N) {
                        const long o = (long)row * NC + col;
                        float v = acc[ct][r];
                        if (BIAS)  v += bias[col];
                        if (ACCUM) v += Out[o];
                        Out[o] = v;
                    }
                }
            }
        }
    }
}

// ---------------------------------------------------------------- scatter
// agg[dst] += hg[src] * dinv[src]*dinv[dst].  One wave per edge, 4 ch/lane.
__global__ __launch_bounds__(256)
void scatter_kernel(const float* __restrict__ hg, const int* __restrict__ src,
                    const int* __restrict__ dst, const float* __restrict__ dinv,
                    float* __restrict__ agg, int E) {
    long tid = (long)blockIdx.x * 256 + threadIdx.x;
    int e = (int)(tid >> 5);
    int g = (int)(tid & 31) * 4;
    if (e >= E) return;
    int s = src[e], d = dst[e];
    float w = dinv[s] * dinv[d];
    float4 v = *(const float4*)(hg + (long)s * D + g);
    float* ap = agg + (long)d * D + g;
    atomicAdd(ap + 0, v.x * w);
    atomicAdd(ap + 1, v.y * w);
    atomicAdd(ap + 2, v.z * w);
    atomicAdd(ap + 3, v.w * w);
}

// ---------------------------------------------------------------- combine + BN stats
// agg (already = scatter + h@Wl) -> h_pre = agg + hg*dinv^2 + bg + bl (in place),
// accumulating per-channel sum / sumsq for BatchNorm. 256 thr = 128 ch x 2 rows.
__global__ __launch_bounds__(256)
void combine_kernel(float* __restrict__ agg, const float* __restrict__ hg,
                    const float* __restrict__ dinv, const float* __restrict__ bg,
                    const float* __restrict__ bl, float* __restrict__ stats, int N) {
    __shared__ float s1[256], s2[256];
    const int t = threadIdx.x;
    const int c = t & 127, rh = t >> 7;
    const int base = blockIdx.x * 32;
    const float bias = bg[c] + bl[c];
    float sum = 0.f, sq = 0.f;
    for (int r = rh; r < 32; r += 2) {
        int n = base + r;
        if (n < N) {
            float dn = dinv[n];
            long o = (long)n * D + c;
            float v = agg[o] + hg[o] * (dn * dn) + bias;
            agg[o] = v;
            sum += v;
            sq  += v * v;
        }
    }
    s1[t] = sum; s2[t] = sq;
    __syncthreads();
    if (rh == 0) {
        sum += s1[t + 128];
        sq  += s2[t + 128];
        atomicAdd(&stats[c], sum);
        atomicAdd(&stats[D + c], sq);
    }
}

// ---------------------------------------------------------------- BN params
__global__ void bnparam_kernel(const float* __restrict__ stats,
                               const float* __restrict__ gamma,
                               const float* __restrict__ beta,
                               float* __restrict__ ab, float invN) {
    int c = threadIdx.x;
    float mu  = stats[c] * invN;
    float var = stats[D + c] * invN - mu * mu;      // biased variance
    float inv = rsqrtf(var + 1e-5f);
    float a = gamma[c] * inv;
    ab[c]     = a;
    ab[D + c] = beta[c] - mu * a;
}

// ---------------------------------------------------------------- BN apply + ReLU + residual accum
__global__ __launch_bounds__(256)
void apply_kernel(const float* __restrict__ hpre, const float* __restrict__ ab,
                  float* __restrict__ h, float* __restrict__ xl, int first, long n4) {
    long i = (long)blockIdx.x * 256 + threadIdx.x;
    if (i >= n4) return;
    int c = (int)((i * 4) & (D - 1));
    float4 v = ((const float4*)hpre)[i];
    float4 r;
    r.x = fmaxf(fmaf(v.x, ab[c + 0], ab[D + c + 0]), 0.f);
    r.y = fmaxf(fmaf(v.y, ab[c + 1], ab[D + c + 1]), 0.f);
    r.z = fmaxf(fmaf(v.z, ab[c + 2], ab[D + c + 2]), 0.f);
    r.w = fmaxf(fmaf(v.w, ab[c + 3], ab[D + c + 3]), 0.f);
    ((float4*)h)[i] = r;
    if (first) {
        ((float4*)xl)[i] = r;
    } else {
        float4 x = ((float4*)xl)[i];
        x.x += r.x; x.y += r.y; x.z += r.z; x.w += r.w;
        ((float4*)xl)[i] = x;
    }
}

// ---------------------------------------------------------------- launch
extern "C" void kernel_launch(void* const* d_in, const int* in_sizes, int n_in,
                              void* d_out, int out_size, void* d_ws, size_t ws_size,
                              hipStream_t stream) {
    const float* x     = (const float*)d_in[0];
    const int*   ei    = (const int*)  d_in[1];
    const float* Wg    = (const float*)d_in[2];
    const float* bg    = (const float*)d_in[3];
    const float* Wl    = (const float*)d_in[4];
    const float* bl    = (const float*)d_in[5];
    const float* gamma = (const float*)d_in[6];
    const float* beta  = (const float*)d_in[7];
    const float* Wp    = (const float*)d_in[8];
    const float* bp    = (const float*)d_in[9];
    float* out = (float*)d_out;
    (void)n_in; (void)out_size; (void)ws_size;

    const int N = in_sizes[0] / D;
    const int E = in_sizes[1] / 2;
    const int* src = ei;
    const int* dst = ei + E;
    const long ND = (long)N * D;

    float* ws = (float*)d_ws;
    size_t off = 0;
    auto alloc = [&](long n) { float* p = ws + off; off += (size_t)((n + 255) & ~255L); return p; };
    float* dinv  = alloc(N);        // deg^{-1/2}
    float* hg    = alloc(ND);       // h @ Wg
    float* agg   = alloc(ND);       // scatter accum -> h_pre (in place)
    float* h     = alloc(ND);       // current features
    float* xl    = alloc(ND);       // x_local accumulator
    float* stats = alloc(2 * D);    // BN sum / sumsq
    float* ab    = alloc(2 * D);    // BN scale / shift

    // degrees -> dinv
    zero_kernel<<<256, 256, 0, stream>>>(dinv, N);
    deg_kernel<<<(E + 255) / 256, 256, 0, stream>>>(dst, dinv, E);
    dinv_kernel<<<(N + 255) / 256, 256, 0, stream>>>(dinv, N);

    const int gblocks = (N + 127) / 128;        // 128 rows per block (2 tiles)
    for (int i = 0; i < LAYERS; ++i) {
        const float* hin = (i == 0) ? x : h;
        gemm_kernel<8, false, false><<<gblocks, 128, 0, stream>>>(
            hin, Wg + (long)i * D * D, hg, nullptr, N);
        zero_kernel<<<2048, 256, 0, stream>>>(agg, ND);
        zero_kernel<<<1, 256, 0, stream>>>(stats, 2 * D);
        scatter_kernel<<<(int)(((long)E * 32 + 255) / 256), 256, 0, stream>>>(
            hg, src, dst, dinv, agg, E);
        gemm_kernel<8, true, false><<<gblocks, 128, 0, stream>>>(      // agg += h @ Wl
            hin, Wl + (long)i * D * D, agg, nullptr, N);
        combine_kernel<<<(N + 31) / 32, 256, 0, stream>>>(
            agg, hg, dinv, bg + i * D, bl + i * D, stats, N);
        bnparam_kernel<<<1, D, 0, stream>>>(
            stats, gamma + i * D, beta + i * D, ab, 1.0f / (float)N);
        apply_kernel<<<(int)((ND / 4 + 255) / 256), 256, 0, stream>>>(
            agg, ab, h, xl, (i == 0) ? 1 : 0, ND / 4);
    }
    // prediction head: out = x_local @ Wp + bp   (128 -> 64 cols)
    gemm_kernel<4, false, true><<<gblocks, 128, 0, stream>>>(xl, Wp, out, bp, N);
}